// Sinkhorn_37065567765151
// MI455X (gfx1250) — compile-verified
//
#include <hip/hip_runtime.h>

// Sinkhorn (exp=False, dummy_row=False) on [B, N=1024, M=1024] f32.
//
// Math: with a = s + eps on the valid nr x nc block, each even iteration is
//   v_j = 1 / sum_{i<nr} a_ij * u_i      (old v cancels exactly)
// and each odd iteration is
//   u_i = 1 / sum_{j<nc} a_ij * v_j      (old u cancels exactly)
// final: out = a_ij * u_i * v_j, with u_i=0 (i>=nr) / v_j=0 (j>=nc) giving
// exact zeros outside the mask with no branches.
//
// Traffic: 10 full-matrix normalization passes (5 GB) collapse to 10 L2-resident
// mat-vecs (4 MB working set per workgroup/batch, 64 batches vs 192 MB L2) plus
// one 512 MB streaming output pass == the HBM floor (~22 us at 23.3 TB/s).
// Output uses nontemporal (TH=NT) b128 stores so it does not evict the
// iterating batches from L2.
//
// Column sums run on the matrix pipe: V_WMMA_F32_16X16X4_F32 with the u-chunk
// replicated into all 16 A-operand rows, accumulating a u-weighted 4x16 tile
// per instruction; D[0][n] (VGPR0, lanes 0-15) holds the 16 column sums.
// Row sums use coalesced b128 loads + wave32 shfl_xor reduction.

typedef float v2f __attribute__((ext_vector_type(2)));
typedef float v4f __attribute__((ext_vector_type(4)));
typedef float v8f __attribute__((ext_vector_type(8)));

#define N_DIM 1024
#define M_DIM 1024
#define SK_EPS 1e-4f
#define SK_ITERS 10

__global__ __launch_bounds__(1024) void sinkhorn_gfx1250_kernel(
    const float* __restrict__ s, const int* __restrict__ nrows,
    const int* __restrict__ ncols, float* __restrict__ out)
{
    __shared__ __align__(16) float u_lds[N_DIM];
    __shared__ __align__(16) float v_lds[M_DIM];

    const int b    = blockIdx.x;
    const int tid  = threadIdx.x;
    const int lane = tid & 31;
    const int wave = tid >> 5;
    const int nr   = nrows[b];
    const int nc   = ncols[b];

    const float* A = s   + (size_t)b * N_DIM * M_DIM;
    float*       O = out + (size_t)b * N_DIM * M_DIM;

    // u_i = 1 inside valid rows, 0 outside (implicit row mask); same for v.
    u_lds[tid] = (tid < nr) ? 1.0f : 0.0f;
    v_lds[tid] = (tid < nc) ? 1.0f : 0.0f;
    __syncthreads();

    const int ngroups = (nc + 15) >> 4;   // 16-column groups to normalize
    const int half    = lane >> 4;        // 0: lanes 0-15, 1: lanes 16-31
    const int nsub    = lane & 15;        // N index within the 16-col group
    const int klo     = half << 1;        // K rows held by this half-wave

    for (int it = 0; it < SK_ITERS / 2; ++it) {
        // ---- even iteration: u-weighted column sums via WMMA, update v ----
        for (int g = wave; g < ngroups; g += 32) {
            const int j0 = g << 4;
            v8f c = {};  // D[m][n] identical over m (A-operand rows replicated)
            for (int k0 = 0; k0 < nr; k0 += 4) {
                const int r0 = k0 + klo;
                const float* p0 = A + (size_t)r0 * M_DIM + j0 + nsub;
                // B operand: 4x16 tile of (s + eps).
                // layout: VGPR0 = rows {k0, k0+2}, VGPR1 = rows {k0+1, k0+3}
                v2f bv;
                bv.x = p0[0]     + SK_EPS;
                bv.y = p0[M_DIM] + SK_EPS;
                // A operand: u[k0..k0+3] replicated, same K mapping as B.
                v2f au;
                au.x = u_lds[r0];
                au.y = u_lds[r0 + 1];
                // Speculative stream-ahead (global_prefetch_b8); OOB prefetch
                // is silently dropped, so no branch needed.
                __builtin_prefetch(p0 + 16 * M_DIM, 0, 1);
                c = __builtin_amdgcn_wmma_f32_16x16x4_f32(
                        false, au, false, bv, (short)0, c, false, false);
            }
            const int j = j0 + nsub;
            if (half == 0 && j < nc) {
                float p = c[0];                  // D[0][nsub] = colsum(j)
                p = (p == 0.0f) ? 1.0f : p;      // reference guard
                v_lds[j] = 1.0f / p;
            }
        }
        __syncthreads();

        // ---- odd iteration: row sums (b128 coalesced) + wave reduce, update u ----
        for (int i = wave; i < nr; i += 32) {
            const float* row = A + (size_t)i * M_DIM;
            float acc = 0.0f;
            // j <= 1020 always (multiple of 4 below nc<=1024), so the float4
            // stays in-bounds; v_lds[j]=0 for j>=nc self-masks the tail.
            for (int j = lane << 2; j < nc; j += 128) {
                const v4f a4 = *(const v4f*)(row + j);
                const v4f vv = *(const v4f*)(v_lds + j);
                acc += (a4.x + SK_EPS) * vv.x;
                acc += (a4.y + SK_EPS) * vv.y;
                acc += (a4.z + SK_EPS) * vv.z;
                acc += (a4.w + SK_EPS) * vv.w;
            }
            for (int off = 16; off > 0; off >>= 1)
                acc += __shfl_xor(acc, off, 32);
            if (lane == 0) {
                float q = (acc == 0.0f) ? 1.0f : acc;  // reference guard
                u_lds[i] = 1.0f / q;
            }
        }
        __syncthreads();
    }

    // ---- final: out = (s + eps) * u_i * v_j (zeros outside mask come from
    //      u/v being 0 there). b128 nontemporal traffic: the 512 MB output
    //      stream must not evict the L2-resident batches still iterating. ----
    const int jf = (tid & 255) << 2;      // this thread's 4 columns
    const int ir = tid >> 8;              // row offset within each 4-row strip
    const v4f vj = *(const v4f*)(v_lds + jf);
    for (int i0 = 0; i0 < N_DIM; i0 += 4) {
        const int i = i0 + ir;
        const float ui = u_lds[i];
        const v4f a4 = __builtin_nontemporal_load(
                           (const v4f*)(A + (size_t)i * M_DIM + jf));
        v4f r;
        r.x = (a4.x + SK_EPS) * ui * vj.x;
        r.y = (a4.y + SK_EPS) * ui * vj.y;
        r.z = (a4.z + SK_EPS) * ui * vj.z;
        r.w = (a4.w + SK_EPS) * ui * vj.w;
        __builtin_nontemporal_store(r, (v4f*)(O + (size_t)i * M_DIM + jf));
    }
}

extern "C" void kernel_launch(void* const* d_in, const int* in_sizes, int n_in,
                              void* d_out, int out_size, void* d_ws, size_t ws_size,
                              hipStream_t stream) {
    const float* s     = (const float*)d_in[0];
    const int*   nrows = (const int*)d_in[1];
    const int*   ncols = (const int*)d_in[2];
    float*       o     = (float*)d_out;
    const int B = in_sizes[1];           // one workgroup per batch matrix
    sinkhorn_gfx1250_kernel<<<dim3(B), dim3(1024), 0, stream>>>(s, nrows, ncols, o);
}